// SRURNNGate_57604101374530
// MI455X (gfx1250) — compile-verified
//
#include <hip/hip_runtime.h>
#include <cstdint>

// ---------------------------------------------------------------------------
// SRU gate recurrence, (B,L,D)=(16,2048,1024) fp32.
// 16384 independent per-(b,d) serial chains of length L.
// 64 blocks (grid = D/256 x B) x 256 threads; one channel per thread.
// 4-deep ring of 16-timestep LDS tiles filled with CDNA5 async
// global->LDS b128 copies (ASYNCcnt, prefetch distance 3 tiles),
// compute uses v_tanh_f32 sigmoid, nontemporal coalesced stores.
// ---------------------------------------------------------------------------

#define AS1 __attribute__((address_space(1)))
#define AS3 __attribute__((address_space(3)))

typedef int v4i __attribute__((vector_size(16)));   // matches builtin param type

#define DCHUNK 256   // channels per block (= blockDim.x)
#define TCHUNK 16    // timesteps per pipeline stage
#define NBUF   4     // ring depth
#define PF     3     // tiles prefetched ahead (= NBUF-1)
#define A_PER_TILE (2 * (TCHUNK / 4))   // async instrs per wave per tile = 8
// LDS: 4 bufs * 2 arrays * 16 * 256 * 4B = 128 KiB (CDNA5: 320 KiB/WGP)

// ---- CDNA5 async global->LDS copy (16B per lane) --------------------------
#if __has_builtin(__builtin_amdgcn_global_load_async_to_lds_b128)
__device__ __forceinline__ void async_cp16(const float* __restrict__ g,
                                           float* __restrict__ l) {
  __builtin_amdgcn_global_load_async_to_lds_b128(
      (AS1 v4i*)g, (AS3 v4i*)l, /*offset=*/0, /*cpol=*/0);
}
#else
__device__ __forceinline__ void async_cp16(const float* __restrict__ g,
                                           float* __restrict__ l) {
  unsigned lds_off = (unsigned)(unsigned long long)(AS3 void*)l;
  asm volatile("global_load_async_to_lds_b128 %0, %1, off"
               :: "v"(lds_off), "v"((unsigned long long)g)
               : "memory");
}
#endif

#if __has_builtin(__builtin_amdgcn_s_wait_asynccnt)
#define WAIT_ASYNC(n) __builtin_amdgcn_s_wait_asynccnt(n)
#else
#define WAIT_ASYNC(n) asm volatile("s_wait_asynccnt %0" :: "n"(n) : "memory")
#endif

// ---- fast sigmoid building blocks -----------------------------------------
#if __has_builtin(__builtin_amdgcn_tanhf)
#define HAVE_TANH 1
__device__ __forceinline__ float fast_tanh(float x) { return __builtin_amdgcn_tanhf(x); }
#elif __has_builtin(__builtin_amdgcn_tanh_f32)
#define HAVE_TANH 1
__device__ __forceinline__ float fast_tanh(float x) { return __builtin_amdgcn_tanh_f32(x); }
#else
#define HAVE_TANH 0
#endif

__global__ __launch_bounds__(DCHUNK) void sru_scan_kernel(
    const float* __restrict__ f, const float* __restrict__ u,
    const float* __restrict__ wfc, float* __restrict__ out,
    int B, int L, int D) {
  __shared__ alignas(16) float sF[NBUF][TCHUNK * DCHUNK];
  __shared__ alignas(16) float sU[NBUF][TCHUNK * DCHUNK];

  const int d0  = blockIdx.x * DCHUNK;   // grid.x = D / DCHUNK (no int div)
  const int b   = blockIdx.y;            // grid.y = B
  const int tid = threadIdx.x;

  const size_t baseBL = (size_t)b * L * D + d0;
  const float* fB = f + baseBL;
  const float* uB = u + baseBL;
  float* outP = out + (size_t)b * D + d0 + tid;   // out[t*B*D + b*D + d]
  const size_t outStride = (size_t)B * D;

  const float w = wfc[d0 + tid];
#if HAVE_TANH
  const float wh = 0.5f * w;                 // sigmoid(x) = 0.5 + 0.5*tanh(x/2)
#else
  const float wl = -1.442695041f * w;        // exp(-x) = exp2(-log2(e)*x)
#endif

  // issue one TCHUNK tile of f and u as async b128 copies (8 per wave)
  auto refill = [&](int t0, int buf) {
#pragma unroll
    for (int k = 0; k < TCHUNK / 4; ++k) {
      const int i   = (tid + k * DCHUNK) * 4;   // float index within tile
      const int row = i >> 8;                   // i / DCHUNK
      const int col = i & (DCHUNK - 1);
      const size_t goff = (size_t)(t0 + row) * D + col;
      async_cp16(fB + goff, &sF[buf][i]);
      async_cp16(uB + goff, &sU[buf][i]);
    }
  };

  float c = 0.0f;
  const int nchunks = L / TCHUNK;

  // prologue: fill PF tiles of the ring
#pragma unroll
  for (int p = 0; p < PF; ++p) {
    if (p < nchunks) refill(p * TCHUNK, p);
  }

  for (int tc = 0; tc < nchunks; ++tc) {
    const int buf = tc & (NBUF - 1);
    if (tc + PF < nchunks) {
      refill((tc + PF) * TCHUNK, (tc + PF) & (NBUF - 1));
      WAIT_ASYNC(PF * A_PER_TILE);   // tile tc landed; PF tiles still in flight
    } else {
      WAIT_ASYNC(0);                 // tail: drain everything (last 3 chunks)
    }
    __syncthreads();

#pragma unroll
    for (int r = 0; r < TCHUNK; ++r) {
      const float ft = sF[buf][r * DCHUNK + tid];
      const float ut = sU[buf][r * DCHUNK + tid];
#if HAVE_TANH
      const float fh = 0.5f * ft;               // off critical chain
      const float y  = fmaf(wh, c, fh);         // chain op 1
      const float T  = fast_tanh(y);            // chain op 2 (v_tanh_f32)
      const float m  = 0.5f * (ut - c);         // parallel with tanh
      c = fmaf(T, m, c + m);                    // chain op 3
#else
      const float fl = -1.442695041f * ft;
      const float y  = fmaf(wl, c, fl);
      const float e  = __builtin_amdgcn_exp2f(y);
      const float s  = __builtin_amdgcn_rcpf(1.0f + e);
      c = fmaf(s, ut - c, c);
#endif
      __builtin_nontemporal_store(c, outP + (size_t)(tc * TCHUNK + r) * outStride);
    }
    __syncthreads();   // protect ring slot before it is overwritten
  }
}

extern "C" void kernel_launch(void* const* d_in, const int* in_sizes, int n_in,
                              void* d_out, int out_size, void* d_ws, size_t ws_size,
                              hipStream_t stream) {
  (void)n_in; (void)out_size; (void)d_ws; (void)ws_size;
  const float* f   = (const float*)d_in[0];
  const float* u   = (const float*)d_in[1];
  const float* wfc = (const float*)d_in[2];
  float* out = (float*)d_out;

  const int D = in_sizes[2];            // 1024
  const int L = 2048;                   // matches reference
  const int B = (int)((size_t)in_sizes[0] / ((size_t)L * D));  // 16

  dim3 grid(D / DCHUNK, B);             // (4, 16) = 64 blocks, 8 wave32s each
  sru_scan_kernel<<<grid, DCHUNK, 0, stream>>>(f, u, wfc, out, B, L, D);
}